// CLUESLoss_34084860461264
// MI455X (gfx1250) — compile-verified
//
#include <hip/hip_runtime.h>
#include <hip/hip_bf16.h>
#include <math.h>

#define N_ROWS 8192
#define N_CLS 8
#define D_EMB 256
#define LS 0.1f
#define ALPHA 0.3f
#define INV_T (1.0f / 0.07f)

typedef float v2f __attribute__((ext_vector_type(2)));
typedef float v8f __attribute__((ext_vector_type(8)));

// workspace layout (float indices):
//   [0,      8192)  invn          1/clip(||emb_i||, 1e-12)
//   [8192,  16384)  rowExp        sum_j exp(sim_ij)          (incl. diagonal)
//   [16384, 24576)  rowMaskSim    sum_{same class, j!=i} sim_ij
//   [24576]         ceSum         sum of focal terms
//   [24584, 24592)  clsCount[8]   (int) class histogram
#define WS_INVN    0
#define WS_ROWEXP  8192
#define WS_ROWMASK 16384
#define WS_CESUM   24576
#define WS_CLSCNT  24584

__global__ void init_k(float* ws) {
    int i = blockIdx.x * 256 + threadIdx.x;
    if (i < N_ROWS) { ws[WS_ROWEXP + i] = 0.0f; ws[WS_ROWMASK + i] = 0.0f; }
    if (i == 0) ws[WS_CESUM] = 0.0f;
    if (i < N_CLS) ((int*)(ws + WS_CLSCNT))[i] = 0;
}

// one wave (32 lanes) per row: inverse norm + class histogram
__global__ void prep_k(const float* __restrict__ emb, const int* __restrict__ tgt,
                       float* ws) {
    int wv = threadIdx.x >> 5;
    int l  = threadIdx.x & 31;
    int row = blockIdx.x * 8 + wv;
    const float4* p = (const float4*)(emb + (size_t)row * D_EMB + l * 8);
    float4 a = p[0], b = p[1];
    float ss = a.x*a.x + a.y*a.y + a.z*a.z + a.w*a.w
             + b.x*b.x + b.y*b.y + b.z*b.z + b.w*b.w;
    for (int off = 16; off >= 1; off >>= 1) ss += __shfl_xor(ss, off, 32);
    if (l == 0) {
        ws[WS_INVN + row] = 1.0f / fmaxf(sqrtf(ss), 1e-12f);
        atomicAdd(((int*)(ws + WS_CLSCNT)) + tgt[row], 1);
    }
}

// focal cross-entropy with label smoothing, one thread per row (C=8)
__global__ void ce_k(const float* __restrict__ logits, const int* __restrict__ tgt,
                     float* ws) {
    int i = blockIdx.x * 256 + threadIdx.x;
    float focal = 0.0f;
    if (i < N_ROWS) {
        const float* lr = logits + (size_t)i * N_CLS;
        float m = lr[0];
        #pragma unroll
        for (int c = 1; c < N_CLS; c++) m = fmaxf(m, lr[c]);
        float se = 0.0f, sum = 0.0f;
        #pragma unroll
        for (int c = 0; c < N_CLS; c++) { se += __expf(lr[c] - m); sum += lr[c]; }
        float lse = m + __logf(se);
        float nll    = lse - lr[tgt[i]];
        float smooth = lse - sum * (1.0f / N_CLS);
        float ce = (1.0f - LS) * nll + LS * smooth;
        float pt = __expf(-ce);
        float om = 1.0f - pt;
        focal = om * om * ce;                      // gamma = 2
    }
    for (int off = 16; off >= 1; off >>= 1) focal += __shfl_xor(focal, off, 32);
    if ((threadIdx.x & 31) == 0) atomicAdd(ws + WS_CESUM, focal);
}

// fused sim = (E E^T) * invn_i * invn_j / T  with streaming exp / masked-sum
// epilogue. Workgroup b owns 16-row block b (A panel in LDS); its 8 waves
// stripe over the 512 j-tiles. WMMA: V_WMMA_F32_16X16X4_F32, K-loop of 64.
__global__ void __launch_bounds__(256) sim_k(const float* __restrict__ emb,
                                             const int* __restrict__ tgt,
                                             float* ws) {
    __shared__ float ldsA[16 * 260];               // pad 256->260 to avoid bank conflicts

    float* invn    = ws + WS_INVN;
    float* rowExp  = ws + WS_ROWEXP;
    float* rowMask = ws + WS_ROWMASK;

    const int i0 = blockIdx.x * 16;
    const int t  = threadIdx.x;

    for (int idx = t; idx < 16 * 256; idx += 256) {
        int r = idx >> 8, c = idx & 255;
        ldsA[r * 260 + c] = emb[(size_t)(i0 + r) * D_EMB + c];
    }
    __syncthreads();

    const int wv   = t >> 5;
    const int l    = t & 31;
    const int q    = l & 15;        // row (A) / col (B) within tile
    const int half = l >> 4;        // K sub-offset selector

    // per-lane row metadata for the 8 C elements (C layout: VGPR e -> M = e + half*8)
    float invnA[8]; int clsA[8];
    #pragma unroll
    for (int e = 0; e < 8; e++) {
        int gi = i0 + e + half * 8;
        invnA[e] = invn[gi];
        clsA[e]  = tgt[gi];
    }

    float accExp[8], accMask[8];
    #pragma unroll
    for (int e = 0; e < 8; e++) { accExp[e] = 0.0f; accMask[e] = 0.0f; }

    for (int jt = wv; jt < N_ROWS / 16; jt += 8) {
        const int j0 = jt * 16;
        const int gj = j0 + q;
        const float* brow = emb + (size_t)gj * D_EMB;
        const float* arow = ldsA + q * 260;

        v8f c = {};
        #pragma unroll 8
        for (int k0 = 0; k0 < D_EMB; k0 += 4) {
            // 32-bit A 16x4 layout: lane<16 holds K=k0,k0+1; lane>=16 holds K=k0+2,k0+3
            v2f a = *(const v2f*)(arow + k0 + half * 2);
            v2f b = *(const v2f*)(brow + k0 + half * 2);
            c = __builtin_amdgcn_wmma_f32_16x16x4_f32(
                    false, a, false, b, (short)0, c, false, false);
        }

        float invnB = invn[gj];
        int   clsB  = tgt[gj];
        #pragma unroll
        for (int e = 0; e < 8; e++) {
            float s = c[e] * invnA[e] * invnB * INV_T;
            accExp[e] += __expf(s);                     // denominator keeps diagonal
            int gi = i0 + e + half * 8;
            if (clsA[e] == clsB && gi != gj) accMask[e] += s;  // mask zeroes diagonal
        }
    }

    // reduce across the 16 lanes sharing each output row, then accumulate globally
    #pragma unroll
    for (int e = 0; e < 8; e++) {
        float ve = accExp[e], vm = accMask[e];
        for (int off = 1; off < 16; off <<= 1) {
            ve += __shfl_xor(ve, off, 32);
            vm += __shfl_xor(vm, off, 32);
        }
        if (q == 0) {
            int gi = i0 + e + half * 8;
            atomicAdd(&rowExp[gi], ve);
            atomicAdd(&rowMask[gi], vm);
        }
    }
}

__global__ void fin_k(const int* __restrict__ tgt, const float* __restrict__ ws,
                      float* out) {
    __shared__ float red[256];
    const float* rowExp  = ws + WS_ROWEXP;
    const float* rowMask = ws + WS_ROWMASK;
    const int*   cls     = (const int*)(ws + WS_CLSCNT);

    float s = 0.0f;
    for (int i = threadIdx.x; i < N_ROWS; i += 256) {
        float L  = __logf(rowExp[i] + 1e-8f);
        int   m  = cls[tgt[i]] - 1;             // positives excluding self
        float mf = (float)m;
        float npos = fmaxf(1.0f, mf);
        s += (L * mf - rowMask[i]) / npos;
    }
    red[threadIdx.x] = s;
    __syncthreads();
    for (int off = 128; off > 0; off >>= 1) {
        if (threadIdx.x < off) red[threadIdx.x] += red[threadIdx.x + off];
        __syncthreads();
    }
    if (threadIdx.x == 0) {
        float contr = red[0] / (float)N_ROWS;
        float ce    = ws[WS_CESUM] / (float)N_ROWS;
        out[0] = ce + ALPHA * contr;
        out[1] = ce;
        out[2] = contr;
    }
}

extern "C" void kernel_launch(void* const* d_in, const int* in_sizes, int n_in,
                              void* d_out, int out_size, void* d_ws, size_t ws_size,
                              hipStream_t stream) {
    const float* logits = (const float*)d_in[0];
    const float* emb    = (const float*)d_in[1];
    const int*   tgt    = (const int*)d_in[2];
    float* out = (float*)d_out;
    float* ws  = (float*)d_ws;

    init_k<<<N_ROWS / 256, 256, 0, stream>>>(ws);
    prep_k<<<N_ROWS / 8, 256, 0, stream>>>(emb, tgt, ws);
    ce_k<<<N_ROWS / 256, 256, 0, stream>>>(logits, tgt, ws);
    sim_k<<<N_ROWS / 16, 256, 0, stream>>>(emb, tgt, ws);
    fin_k<<<1, 256, 0, stream>>>(tgt, ws, out);
}